// cofeature_fast_62010737820065
// MI455X (gfx1250) — compile-verified
//
#include <hip/hip_runtime.h>

typedef __attribute__((ext_vector_type(2))) float v2f;
typedef __attribute__((ext_vector_type(8))) float v8f;

#define NB   16
#define NC   192
#define NH   64
#define NW   64
#define NK   (NH * NW)          // 4096
#define CBLK 32                 // c-rows per wave (2 A fragments)
#define NCB  (NC / CBLK)        // 6 c-blocks
#define NDT  (NC / 16)          // 12 d-tiles
#define NOFF 5
#define CC   (NC * NC)          // 36864

// D = A*B + C, fp32 16x16x4
#define WM(A_, B_, C_) \
    __builtin_amdgcn_wmma_f32_16x16x4_f32(false, (A_), false, (B_), (short)0, (C_), false, false)

// 10 rank-4 updates: 2 A fragments x 5 offsets, all-distinct accumulators (no RAW)
#define WMMA10(a0_, a1_, b0_, b1_, b2_, b3_, b4_) do {            \
    accA0 = WM(a0_, b0_, accA0); accB0 = WM(a1_, b0_, accB0);     \
    accA1 = WM(a0_, b1_, accA1); accB1 = WM(a1_, b1_, accB1);     \
    accA2 = WM(a0_, b2_, accA2); accB2 = WM(a1_, b2_, accB2);     \
    accA3 = WM(a0_, b3_, accA3); accB3 = WM(a1_, b3_, accB3);     \
    accA4 = WM(a0_, b4_, accA4); accB4 = WM(a1_, b4_, accB4);     \
} while (0)

// ---- zero the per-(b,offset) sum-of-squares accumulators in workspace ----
__global__ void cofe_zero_ws(float* ws) {
    int i = threadIdx.x;
    if (i < NB * NOFF) ws[i] = 0.0f;
}

// ---- per-offset tile epilogue: store 16x16 tile + wave-reduced sum of squares ----
__device__ __forceinline__ void tile_epilogue(v8f acc, int o, int b, int cb, int db,
                                              int m_off, int n, int lane,
                                              float* __restrict__ out,
                                              float* __restrict__ ss) {
    size_t base = (size_t)(b * NOFF + o) * CC + (size_t)(cb + m_off) * NC + (size_t)(db + n);
    float s = 0.0f;
#pragma unroll
    for (int i = 0; i < 8; ++i) {
        float v = acc[i];
        out[base + (size_t)i * NC] = v;
        s += v * v;
    }
#pragma unroll
    for (int off = 16; off > 0; off >>= 1)
        s += __shfl_xor(s, off, 32);
    if (lane == 0)
        atomicAdd(&ss[b * NOFF + o], s);
}

// ---- main GEMM kernel: 1 wave = one 32x16 (c,d) tile, all 5 offsets ----
// G[c,d] = sum_k x[b,c,k] * x_shift[b,d,k]  (5 shifted variants, edge-clamped)
__global__ __launch_bounds__(256) void cofe_wmma(const float* __restrict__ x,
                                                 float* __restrict__ out,
                                                 float* __restrict__ ss) {
    const int wave   = (blockIdx.x << 3) + (threadIdx.x >> 5);
    const int lane   = threadIdx.x & 31;
    const int lane15 = lane & 15;
    const int kk     = (lane >> 4) << 1;      // 0 (lanes 0-15) or 2 (lanes 16-31)

    const int b  = wave / (NCB * NDT);
    const int rm = wave % (NCB * NDT);
    const int cb = (rm / NDT) * CBLK;         // center (row) block base, 32 rows
    const int db = (rm % NDT) * 16;           // side (col) tile base, 16 rows

    const float* __restrict__ xb = x + (size_t)b * NC * NK;  // wave-uniform base
    const int xrow = (cb + lane15) * NK;      // A rows (lane-private, 32-bit idx)
    const int yrow = (db + lane15) * NK;      // B rows (lane-private, 32-bit idx)

    v8f accA0 = {}, accA1 = {}, accA2 = {}, accA3 = {}, accA4 = {};  // c rows 0..15
    v8f accB0 = {}, accB1 = {}, accB2 = {}, accB3 = {}, accB4 = {};  // c rows 16..31

    for (int h = 0; h < NH; ++h) {
        const int hm  = (h > 0) ? (h - 1) : 0;        // clamp(h-1): per-row, hoisted
        const int rm1 = yrow + (hm << 6);             // shifted-row base (dy=-1)
        const int r0  = yrow + (h  << 6);             // same-row base (dy=0)
        const int xr  = xrow + (h  << 6);             // A-row base

        // Generic (edge-clamped) k-step; only used for w-edges of each image row.
        auto clamped_step = [&](int w) {
            const int wm  = (w > 0)  ? (w - 1) : 0;   // clamp(w-1)
            const int wp2 = (w < 62) ? (w + 2) : 63;  // clamp(w+2)
            v2f a0 = *(const v2f*)(xb + xr + w);
            v2f a1 = *(const v2f*)(xb + xr + 16 * NK + w);
            v2f b0, b1, b2, b3, b4;
            b0.x = xb[rm1 + wm];    b0.y = xb[rm1 + w];        // (-1,-1)
            b1   = *(const v2f*)(xb + rm1 + w);                // (-1, 0)
            b2.x = xb[rm1 + w + 1]; b2.y = xb[rm1 + wp2];      // (-1,+1)
            b3.x = xb[r0 + wm];     b3.y = xb[r0 + w];         // ( 0,-1)
            b4   = *(const v2f*)(xb + r0 + w);                 // ( 0, 0)
            WMMA10(a0, a1, b0, b1, b2, b3, b4);
        };

        clamped_step(kk);                             // first k-step of row (w=0|2)

        // Interior k-steps: w in [4+kk, 56+kk]; no clamps can fire, so all five
        // B fragments come from two contiguous runs:
        //   row h-1: cols w-1..w+2  (4 floats)  -> b0,b1,b2
        //   row h  : cols w-1..w+1  (3 floats)  -> b3,b4
        for (int wi = 1; wi < 15; ++wi) {
            const int w = (wi << 2) + kk;
            v2f a0 = *(const v2f*)(xb + xr + w);
            v2f a1 = *(const v2f*)(xb + xr + 16 * NK + w);

            const float* __restrict__ pu = xb + rm1 + w - 1;
            const float* __restrict__ pc = xb + r0  + w - 1;
            float p0 = pu[0], p1 = pu[1], p2 = pu[2], p3 = pu[3];
            float q0 = pc[0], q1 = pc[1], q2 = pc[2];

            v2f b0 = {p0, p1};   // (-1,-1)
            v2f b1 = {p1, p2};   // (-1, 0)
            v2f b2 = {p2, p3};   // (-1,+1)
            v2f b3 = {q0, q1};   // ( 0,-1)
            v2f b4 = {q1, q2};   // ( 0, 0)
            WMMA10(a0, a1, b0, b1, b2, b3, b4);
        }

        clamped_step(60 + kk);                        // last k-step of row (w=60|62)
    }

    // C/D layout: VGPR i holds (M = i + 8*(lane>=16), N = lane&15)
    const int m_off = (lane >> 4) << 3;
    tile_epilogue(accA0, 0, b, cb,      db, m_off, lane15, lane, out, ss);
    tile_epilogue(accB0, 0, b, cb + 16, db, m_off, lane15, lane, out, ss);
    tile_epilogue(accA1, 1, b, cb,      db, m_off, lane15, lane, out, ss);
    tile_epilogue(accB1, 1, b, cb + 16, db, m_off, lane15, lane, out, ss);
    tile_epilogue(accA2, 2, b, cb,      db, m_off, lane15, lane, out, ss);
    tile_epilogue(accB2, 2, b, cb + 16, db, m_off, lane15, lane, out, ss);
    tile_epilogue(accA3, 3, b, cb,      db, m_off, lane15, lane, out, ss);
    tile_epilogue(accB3, 3, b, cb + 16, db, m_off, lane15, lane, out, ss);
    tile_epilogue(accA4, 4, b, cb,      db, m_off, lane15, lane, out, ss);
    tile_epilogue(accB4, 4, b, cb + 16, db, m_off, lane15, lane, out, ss);
}

// ---- normalization: out[i] /= max(sqrt(ss[b*5+o]), 1e-12) ----
__global__ __launch_bounds__(256) void cofe_normalize(float* __restrict__ out,
                                                      const float* __restrict__ ss) {
    size_t i = (size_t)blockIdx.x * 256 + threadIdx.x;
    size_t grp = i / CC;                      // b*5 + o
    float nrm = sqrtf(ss[grp]);
    nrm = fmaxf(nrm, 1e-12f);
    out[i] = out[i] / nrm;
}

extern "C" void kernel_launch(void* const* d_in, const int* in_sizes, int n_in,
                              void* d_out, int out_size, void* d_ws, size_t ws_size,
                              hipStream_t stream) {
    const float* x  = (const float*)d_in[0];
    float* out      = (float*)d_out;
    float* ss       = (float*)d_ws;           // 80 floats of scratch

    cofe_zero_ws<<<1, 128, 0, stream>>>(ss);

    // 16 batches * 6 c-blocks * 12 d-tiles = 1152 waves; 8 waves / 256-thread block
    cofe_wmma<<<(NB * NCB * NDT) / 8, 256, 0, stream>>>(x, out, ss);

    // 16*5*36864 = 2,949,120 elements / 256 = 11,520 blocks (exact)
    cofe_normalize<<<(NB * NOFF * CC) / 256, 256, 0, stream>>>(out, ss);
}